// SMPL_67912022884411
// MI455X (gfx1250) — compile-verified
//
#include <hip/hip_runtime.h>

typedef float v2f __attribute__((ext_vector_type(2)));
typedef float v4f __attribute__((ext_vector_type(4)));
typedef float v8f __attribute__((ext_vector_type(8)));

#define NB 512
#define NV 6890
#define NJ 24
#define N3V 20670   // 3*NV

__constant__ int c_par[24] = {0,0,0,0,1,2,3,4,5,6,7,8,9,9,9,12,13,14,16,17,18,19,20,21};

// ---------------- beta zero-pad to K=12 ----------------
__global__ void k_beta_pad(const float* __restrict__ beta, float* __restrict__ bp) {
    int i = blockIdx.x * blockDim.x + threadIdx.x;
    if (i >= NB * 12) return;
    int b = i / 12, k = i % 12;
    bp[i] = (k < 10) ? beta[b * 10 + k] : 0.f;
}

// ---------------- Rodrigues -> Rs and padded pose_feature [512,208] ----------------
__global__ void k_rod(const float* __restrict__ theta, float* __restrict__ Rs,
                      float* __restrict__ pf) {
    int idx = blockIdx.x * blockDim.x + threadIdx.x;
    if (idx >= NB * NJ) return;
    int b = idx / NJ, j = idx % NJ;
    float rx = theta[b * 72 + j * 3 + 0];
    float ry = theta[b * 72 + j * 3 + 1];
    float rz = theta[b * 72 + j * 3 + 2];
    float ax = rx + 1e-8f, ay = ry + 1e-8f, az = rz + 1e-8f;
    float angle = sqrtf(ax * ax + ay * ay + az * az);
    float inv = 1.f / angle;
    float x = rx * inv, y = ry * inv, z = rz * inv;
    float cn = cosf(angle), sn = sinf(angle), oc = 1.f - cn;
    float R[9];
    R[0] = 1.f + oc * (-(z * z + y * y));
    R[1] = sn * (-z) + oc * (x * y);
    R[2] = sn * ( y) + oc * (x * z);
    R[3] = sn * ( z) + oc * (x * y);
    R[4] = 1.f + oc * (-(z * z + x * x));
    R[5] = sn * (-x) + oc * (y * z);
    R[6] = sn * (-y) + oc * (x * z);
    R[7] = sn * ( x) + oc * (y * z);
    R[8] = 1.f + oc * (-(x * x + y * y));
#pragma unroll
    for (int e = 0; e < 9; ++e) Rs[idx * 9 + e] = R[e];
    if (j > 0) {
        float* p = pf + b * 208 + (j - 1) * 9;
#pragma unroll
        for (int e = 0; e < 9; ++e) p[e] = R[e] - ((e % 4 == 0) ? 1.f : 0.f);
    } else {
        pf[b * 208 + 207] = 0.f;   // K-pad
    }
}

// ---------------- J_template = Jr^T @ v_template ----------------
__global__ void k_jtemplate(const float* __restrict__ Jr, const float* __restrict__ vt,
                            float* __restrict__ Jt) {
    int t = threadIdx.x;
    if (t >= 72) return;
    int j = t / 3, c = t % 3;
    float acc = 0.f;
    for (int v = 0; v < NV; ++v) acc += Jr[v * NJ + j] * vt[v * 3 + c];
    Jt[t] = acc;
}

// ---------------- f32 WMMA GEMM: out[M,N] = A[M,Kp] @ B[Kt,N] + addend ----------------
// A must be zero-padded in K up to Kp (multiple of 4). All inner-loop loads are
// unconditional (branch-free): column clamped for address safety (edge lanes feed
// only unstored C columns), K-tail rows clamped (A zeros annihilate the products).
// MODE 0: addend[n] (broadcast over rows);  MODE 1: addend[m*N+n] (in-place ok)
template <int MODE>
__global__ void wmma_gemm(const float* __restrict__ A, const float* __restrict__ B,
                          const float* __restrict__ add, float* __restrict__ out,
                          int M, int N, int Kt, int Kp) {
    const int lane = threadIdx.x & 31;
    const int wave = threadIdx.x >> 5;
    const int ntn  = (N + 15) >> 4;
    const int ntm  = M >> 4;                 // M is a multiple of 16 here (512)
    const int tile = blockIdx.x * (blockDim.x >> 5) + wave;
    if (tile >= ntm * ntn) return;
    const int mt = tile / ntn, nt = tile % ntn;
    const int mrow  = (mt << 4) + (lane & 15);   // A-matrix row for this lane
    const int ncol  = (nt << 4) + (lane & 15);   // B/C column for this lane
    const int ncl   = (ncol < N) ? ncol : (N - 1);
    const int khalf = (lane >> 4) << 1;          // 0 for lanes 0-15, 2 for 16-31

    const float* arow = A + (long)mrow * Kp + khalf;     // 8B-aligned (Kp even)
    const float* bp   = B + (long)khalf * N + ncl;
    const long bstep  = (long)4 * N;
    const int  Ktq    = Kt & ~3;

    v8f c = {};
    int k0 = 0;
    for (; k0 + 4 <= Ktq; k0 += 4) {
        v2f a = *(const v2f*)arow;
        v2f b;
        b.x = bp[0];
        b.y = bp[N];
        c = __builtin_amdgcn_wmma_f32_16x16x4_f32(false, a, false, b,
                                                  (short)0, c, false, false);
        arow += 4;
        bp   += bstep;
    }
    if (k0 < Kp) {   // tail quad
        int kb0 = k0 + khalf;     if (kb0 > Kt - 1) kb0 = Kt - 1;
        int kb1 = k0 + khalf + 1; if (kb1 > Kt - 1) kb1 = Kt - 1;
        v2f a = *(const v2f*)arow;
        v2f b;
        b.x = B[(long)kb0 * N + ncl];
        b.y = B[(long)kb1 * N + ncl];
        c = __builtin_amdgcn_wmma_f32_16x16x4_f32(false, a, false, b,
                                                  (short)0, c, false, false);
    }
    if (ncol < N) {
        const int mbase = (mt << 4) + ((lane >> 4) << 3);  // rows r / r+8 split
#pragma unroll
        for (int r = 0; r < 8; ++r) {
            long oi = (long)(mbase + r) * N + ncol;
            float av = (MODE == 0) ? add[ncol] : add[oi];
            out[oi] = c[r] + av;
        }
    }
}

// ---------------- Jnt (LDS-tiled reduction) + delta_J ----------------
__global__ void k_jnt(const float* __restrict__ vshaped, const float* __restrict__ Jr,
                      const float* __restrict__ Jt, float* __restrict__ Jnt,
                      float* __restrict__ dJ) {
    __shared__ float svs[128 * 3];
    __shared__ float sjr[128 * 24];
    int b = blockIdx.x, t = threadIdx.x;
    int j = t / 3, cc = t % 3;
    float acc = 0.f;
    for (int v0 = 0; v0 < NV; v0 += 128) {
        int nv = (NV - v0 < 128) ? (NV - v0) : 128;
        for (int i = t; i < nv * 3;  i += 128) svs[i] = vshaped[(long)b * N3V + v0 * 3 + i];
        for (int i = t; i < nv * 24; i += 128) sjr[i] = Jr[v0 * 24 + i];
        __syncthreads();
        if (t < 72)
            for (int v = 0; v < nv; ++v) acc += svs[v * 3 + cc] * sjr[v * 24 + j];
        __syncthreads();
    }
    if (t < 72) {
        Jnt[b * 72 + t] = acc;
        dJ [b * 72 + t] = acc - Jt[t];
    }
}

// ---------------- kinematic chain per batch: A and pinv(A[:,0]) ----------------
__global__ void k_chain(const float* __restrict__ Rs, const float* __restrict__ Jnt,
                        float* __restrict__ Aout, float* __restrict__ invOut) {
    __shared__ float tm[24][16];
    __shared__ float ch[24][16];
    __shared__ float jl[72];
    __shared__ float ta[3];
    int b = blockIdx.x, t = threadIdx.x;      // blockDim = 32
    for (int i = t; i < 72; i += 32) jl[i] = Jnt[b * 72 + i];
    __syncthreads();
    if (t < 24) {
        int j = t, par = c_par[j];
        float r0 = jl[j * 3 + 0], r1 = jl[j * 3 + 1], r2 = jl[j * 3 + 2];
        if (j > 0) { r0 -= jl[par * 3 + 0]; r1 -= jl[par * 3 + 1]; r2 -= jl[par * 3 + 2]; }
        const float* R = Rs + (b * NJ + j) * 9;
        tm[j][0] = R[0]; tm[j][1] = R[1]; tm[j][2]  = R[2]; tm[j][3]  = r0;
        tm[j][4] = R[3]; tm[j][5] = R[4]; tm[j][6]  = R[5]; tm[j][7]  = r1;
        tm[j][8] = R[6]; tm[j][9] = R[7]; tm[j][10] = R[8]; tm[j][11] = r2;
        tm[j][12] = 0.f; tm[j][13] = 0.f; tm[j][14] = 0.f; tm[j][15] = 1.f;
    }
    __syncthreads();
    if (t < 16) ch[0][t] = tm[0][t];
    __syncthreads();
    for (int i = 1; i < 24; ++i) {
        if (t < 16) {
            int p = t >> 2, q = t & 3, par = c_par[i];
            float acc = 0.f;
#pragma unroll
            for (int k = 0; k < 4; ++k) acc += ch[par][p * 4 + k] * tm[i][k * 4 + q];
            ch[i][t] = acc;
        }
        __syncthreads();
    }
    // A = transforms - pad(transforms @ [J,0])
    for (int e = t; e < 384; e += 32) {
        int i = e >> 4, pq = e & 15, p = pq >> 2, q = pq & 3;
        float val = ch[i][pq];
        if (q == 3)
            val = ch[i][p * 4 + 3] - (ch[i][p * 4 + 0] * jl[i * 3 + 0] +
                                      ch[i][p * 4 + 1] * jl[i * 3 + 1] +
                                      ch[i][p * 4 + 2] * jl[i * 3 + 2]);
        Aout[(long)b * 384 + e] = val;
    }
    if (t < 3)
        ta[t] = ch[0][t * 4 + 3] - (ch[0][t * 4 + 0] * jl[0] +
                                    ch[0][t * 4 + 1] * jl[1] +
                                    ch[0][t * 4 + 2] * jl[2]);
    __syncthreads();
    // rigid inverse == pinv for the root transform
    if (t < 16) {
        int p = t >> 2, q = t & 3;
        float val;
        if (p < 3 && q < 3)      val = ch[0][q * 4 + p];                         // R^T
        else if (p < 3)          val = -(ch[0][0 * 4 + p] * ta[0] +
                                         ch[0][1 * 4 + p] * ta[1] +
                                         ch[0][2 * 4 + p] * ta[2]);              // -R^T t
        else                     val = (q == 3) ? 1.f : 0.f;
        invOut[b * 16 + t] = val;
    }
}

// ---------------- skinning: verts + W broadcast (non-temporal) ----------------
__global__ void k_skin(const float* __restrict__ w, const float* __restrict__ Abuf,
                       const float* __restrict__ vposed, float* __restrict__ verts,
                       float* __restrict__ Wout) {
    __shared__ float As[24 * 16];
    int b = blockIdx.y;
    for (int i = threadIdx.x; i < 384; i += blockDim.x) As[i] = Abuf[(long)b * 384 + i];
    __syncthreads();
    int v = blockIdx.x * blockDim.x + threadIdx.x;
    if (v >= NV) return;
    long vo = (long)b * N3V + v * 3;
    float x = vposed[vo + 0], y = vposed[vo + 1], z = vposed[vo + 2];
    v4f wv[6];
    const v4f* wrow = (const v4f*)(w + (long)v * 24);
#pragma unroll
    for (int i = 0; i < 6; ++i) wv[i] = wrow[i];
    float o0 = 0.f, o1 = 0.f, o2 = 0.f;
#pragma unroll
    for (int j = 0; j < 24; ++j) {
        float wj = wv[j >> 2][j & 3];
        const float* Aj = As + j * 16;
        o0 += wj * (Aj[0] * x + Aj[1] * y + Aj[2]  * z + Aj[3]);
        o1 += wj * (Aj[4] * x + Aj[5] * y + Aj[6]  * z + Aj[7]);
        o2 += wj * (Aj[8] * x + Aj[9] * y + Aj[10] * z + Aj[11]);
    }
    verts[vo + 0] = o0; verts[vo + 1] = o1; verts[vo + 2] = o2;
    v4f* wout = (v4f*)(Wout + (long)b * (NV * 24) + (long)v * 24);
#pragma unroll
    for (int i = 0; i < 6; ++i) __builtin_nontemporal_store(wv[i], wout + i);
}

extern "C" void kernel_launch(void* const* d_in, const int* in_sizes, int n_in,
                              void* d_out, int out_size, void* d_ws, size_t ws_size,
                              hipStream_t stream) {
    const float* beta       = (const float*)d_in[0];
    const float* theta      = (const float*)d_in[1];
    const float* v_template = (const float*)d_in[2];
    const float* shapedirs  = (const float*)d_in[3];
    const float* Jreg       = (const float*)d_in[4];
    const float* posedirs   = (const float*)d_in[5];
    const float* weights    = (const float*)d_in[6];

    float* out     = (float*)d_out;
    float* v_posed = out;                    // [512,6890,3]
    float* verts   = out + 10583040;         // [512,6890,3]
    float* dJ      = out + 21166080;         // [512,24,3]
    float* Aout    = out + 21202944;         // [512,24,4,4]
    float* Wout    = out + 21399552;         // [512,6890,24]
    float* invG    = out + 106063872;        // [512,4,4]

    float* ws   = (float*)d_ws;
    float* Rs   = ws;                        // 512*24*9  = 110592
    float* pf   = ws + 110592;               // 512*208   = 106496
    float* bpad = ws + 217088;               // 512*12    = 6144
    float* Jnt  = ws + 223232;               // 512*24*3  = 36864
    float* Jt   = ws + 260096;               // 72

    k_beta_pad<<<24, 256, 0, stream>>>(beta, bpad);
    k_rod<<<96, 128, 0, stream>>>(theta, Rs, pf);
    k_jtemplate<<<1, 96, 0, stream>>>(Jreg, v_template, Jt);

    // v_shaped = beta_pad @ shapedirs + v_template  -> stored in v_posed slot
    const int ntiles = (512 / 16) * ((N3V + 15) / 16);   // 32*1292 = 41344
    wmma_gemm<0><<<(ntiles + 7) / 8, 256, 0, stream>>>(bpad, shapedirs, v_template,
                                                       v_posed, 512, N3V, 10, 12);
    k_jnt<<<512, 128, 0, stream>>>(v_posed, Jreg, Jt, Jnt, dJ);
    k_chain<<<512, 32, 0, stream>>>(Rs, Jnt, Aout, invG);

    // v_posed = pose_feature @ posedirs + v_shaped  (in-place on v_posed slot)
    wmma_gemm<1><<<(ntiles + 7) / 8, 256, 0, stream>>>(pf, posedirs, v_posed,
                                                       v_posed, 512, N3V, 207, 208);
    k_skin<<<dim3(27, 512), 256, 0, stream>>>(weights, Aout, v_posed, verts, Wout);
}